// MoEGate_10187662426952
// MI455X (gfx1250) — compile-verified
//
#include <hip/hip_runtime.h>
#include <hip/hip_bf16.h>
#include <math.h>

typedef float v2f __attribute__((ext_vector_type(2)));
typedef float v4f __attribute__((ext_vector_type(4)));
typedef float v8f __attribute__((ext_vector_type(8)));

#define TOPK 8
#define NEXPERT 64
#define TOK_PER_BLK 32   // 2 x 16-token WMMA tiles
#define THREADS 128      // 4 wave32 -> 4 x 16-expert slices

// Each block: 32 tokens x 64 experts of gate logits via V_WMMA_F32_16X16X4_F32,
// then per-token softmax + top-8 epilogue out of LDS.
// H is a template parameter so the K-loop has a compile-time trip count:
// no remainder path, immediate-offset addressing, deeper unroll.
template <int H>
__global__ __launch_bounds__(THREADS) void MoEGate_10187662426952_kernel(
    const float* __restrict__ x,   // [T, H] hidden states
    const float* __restrict__ w,   // [64, H] gate weight
    int*   __restrict__ out_idx,   // [T, 8]
    float* __restrict__ out_w,     // [T, 8]
    int*   __restrict__ out_row,   // [T, 8]
    int T)
{
    __shared__ float lds[TOK_PER_BLK * NEXPERT];  // 8 KB logits tile

    const int wave = threadIdx.x >> 5;     // 0..3 -> expert slice
    const int lane = threadIdx.x & 31;
    const int half = lane >> 4;            // 0/1: lane half
    const int lr   = lane & 15;            // row within half
    const int tb   = blockIdx.x * TOK_PER_BLK;
    const int e0   = wave * 16;

    // A fragment rows: token tb+lr (tile0) / tb+16+lr (tile1); B rows: expert e0+lr.
    // Fold the per-lane-half K offset into the base pointer once.
    const float* xr0 = x + (size_t)(tb + lr) * H + 4 * half;
    const float* xr1 = x + (size_t)(tb + 16 + lr) * H + 4 * half;
    const float* wr  = w + (size_t)(e0 + lr) * H + 4 * half;

    v8f acc0 = {};   // C tile for tokens tb..tb+15
    v8f acc1 = {};   // C tile for tokens tb+16..tb+31

    // Consume 8 K-values per iteration. Legal K-permutation: lane half 0 covers
    // logical k0..k0+3, half 1 covers k0+4..k0+7 (A and B agree), so each lane's
    // fragment data for two WMMA steps is one contiguous 16-byte load.
    #pragma unroll 4
    for (int k0 = 0; k0 < H; k0 += 8) {
        v4f b4 = *(const v4f*)(wr  + k0);   // B shared by both token tiles
        v4f a4 = *(const v4f*)(xr0 + k0);
        v4f c4 = *(const v4f*)(xr1 + k0);

        v2f bLo = b4.xy, bHi = b4.zw;
        v2f aLo = a4.xy, aHi = a4.zw;
        v2f cLo = c4.xy, cHi = c4.zw;

        acc0 = __builtin_amdgcn_wmma_f32_16x16x4_f32(false, aLo, false, bLo,
                                                     (short)0, acc0, false, false);
        acc1 = __builtin_amdgcn_wmma_f32_16x16x4_f32(false, cLo, false, bLo,
                                                     (short)0, acc1, false, false);
        acc0 = __builtin_amdgcn_wmma_f32_16x16x4_f32(false, aHi, false, bHi,
                                                     (short)0, acc0, false, false);
        acc1 = __builtin_amdgcn_wmma_f32_16x16x4_f32(false, cHi, false, bHi,
                                                     (short)0, acc1, false, false);
    }

    // C layout: element r of acc is logits[token = r + 8*half][expert = e0 + lr].
    #pragma unroll
    for (int r = 0; r < 8; ++r) {
        lds[(r + 8 * half) * NEXPERT + e0 + lr]        = acc0[r];
        lds[(16 + r + 8 * half) * NEXPERT + e0 + lr]   = acc1[r];
    }
    __syncthreads();

    // ---- Epilogue: softmax + top-8, one thread per token (threads 0..31) ----
    if (threadIdx.x < TOK_PER_BLK) {
        const int t   = threadIdx.x;
        float* row    = lds + t * NEXPERT;
        const int rot = (2 * t) & (NEXPERT - 1);   // bank-conflict-free scan

        float m = -INFINITY;
        for (int e = 0; e < NEXPERT; ++e)
            m = fmaxf(m, row[(e + rot) & (NEXPERT - 1)]);

        float s = 0.f;
        for (int e = 0; e < NEXPERT; ++e) {
            const int ei = (e + rot) & (NEXPERT - 1);
            const float p = __expf(row[ei] - m);
            row[ei] = p;                            // thread-private row, safe
            s += p;
        }
        const float inv_s = 1.0f / s;

        float topw[TOPK];
        int   topi[TOPK];
        float wsum = 0.f;
        #pragma unroll
        for (int k = 0; k < TOPK; ++k) {
            float best = -INFINITY;
            int   bi   = NEXPERT;
            for (int e = 0; e < NEXPERT; ++e) {
                const int ei = (e + rot) & (NEXPERT - 1);
                const float v = row[ei];
                // tie-break: lowest expert index first (matches jax.lax.top_k)
                if (v > best || (v == best && ei < bi)) { best = v; bi = ei; }
            }
            row[bi] = -INFINITY;                    // exclude from next pass
            const float sw = best * inv_s;          // softmax score
            topw[k] = sw;
            topi[k] = bi;
            wsum += sw;
        }

        const float denom = wsum + 1e-20f;
        const int gt = tb + t;
        #pragma unroll
        for (int k = 0; k < TOPK; ++k) {
            out_idx[(size_t)gt * TOPK + k] = topi[k];
            out_w  [(size_t)gt * TOPK + k] = topw[k] / denom;
            out_row[(size_t)gt * TOPK + k] = k * T + gt;   // column-major token ids
        }
    }
}

// Generic-H fallback (same algorithm, runtime trip count).
__global__ __launch_bounds__(THREADS) void MoEGate_10187662426952_kernel_dyn(
    const float* __restrict__ x, const float* __restrict__ w,
    int* __restrict__ out_idx, float* __restrict__ out_w,
    int* __restrict__ out_row, int T, int H)
{
    __shared__ float lds[TOK_PER_BLK * NEXPERT];
    const int wave = threadIdx.x >> 5;
    const int lane = threadIdx.x & 31;
    const int half = lane >> 4;
    const int lr   = lane & 15;
    const int tb   = blockIdx.x * TOK_PER_BLK;
    const int e0   = wave * 16;

    const float* xr0 = x + (size_t)(tb + lr) * H + 4 * half;
    const float* xr1 = x + (size_t)(tb + 16 + lr) * H + 4 * half;
    const float* wr  = w + (size_t)(e0 + lr) * H + 4 * half;

    v8f acc0 = {}; v8f acc1 = {};
    for (int k0 = 0; k0 < H; k0 += 8) {
        v4f b4 = *(const v4f*)(wr  + k0);
        v4f a4 = *(const v4f*)(xr0 + k0);
        v4f c4 = *(const v4f*)(xr1 + k0);
        acc0 = __builtin_amdgcn_wmma_f32_16x16x4_f32(false, a4.xy, false, b4.xy,
                                                     (short)0, acc0, false, false);
        acc1 = __builtin_amdgcn_wmma_f32_16x16x4_f32(false, c4.xy, false, b4.xy,
                                                     (short)0, acc1, false, false);
        acc0 = __builtin_amdgcn_wmma_f32_16x16x4_f32(false, a4.zw, false, b4.zw,
                                                     (short)0, acc0, false, false);
        acc1 = __builtin_amdgcn_wmma_f32_16x16x4_f32(false, c4.zw, false, b4.zw,
                                                     (short)0, acc1, false, false);
    }
    #pragma unroll
    for (int r = 0; r < 8; ++r) {
        lds[(r + 8 * half) * NEXPERT + e0 + lr]      = acc0[r];
        lds[(16 + r + 8 * half) * NEXPERT + e0 + lr] = acc1[r];
    }
    __syncthreads();

    if (threadIdx.x < TOK_PER_BLK) {
        const int t = threadIdx.x;
        float* row = lds + t * NEXPERT;
        const int rot = (2 * t) & (NEXPERT - 1);
        float m = -INFINITY;
        for (int e = 0; e < NEXPERT; ++e)
            m = fmaxf(m, row[(e + rot) & (NEXPERT - 1)]);
        float s = 0.f;
        for (int e = 0; e < NEXPERT; ++e) {
            const int ei = (e + rot) & (NEXPERT - 1);
            const float p = __expf(row[ei] - m);
            row[ei] = p; s += p;
        }
        const float inv_s = 1.0f / s;
        float topw[TOPK]; int topi[TOPK]; float wsum = 0.f;
        #pragma unroll
        for (int k = 0; k < TOPK; ++k) {
            float best = -INFINITY; int bi = NEXPERT;
            for (int e = 0; e < NEXPERT; ++e) {
                const int ei = (e + rot) & (NEXPERT - 1);
                const float v = row[ei];
                if (v > best || (v == best && ei < bi)) { best = v; bi = ei; }
            }
            row[bi] = -INFINITY;
            const float sw = best * inv_s;
            topw[k] = sw; topi[k] = bi; wsum += sw;
        }
        const float denom = wsum + 1e-20f;
        const int gt = tb + t;
        #pragma unroll
        for (int k = 0; k < TOPK; ++k) {
            out_idx[(size_t)gt * TOPK + k] = topi[k];
            out_w  [(size_t)gt * TOPK + k] = topw[k] / denom;
            out_row[(size_t)gt * TOPK + k] = k * T + gt;
        }
    }
}

extern "C" void kernel_launch(void* const* d_in, const int* in_sizes, int n_in,
                              void* d_out, int out_size, void* d_ws, size_t ws_size,
                              hipStream_t stream) {
    const float* x = (const float*)d_in[0];   // hidden_states [4,4096,2048] f32
    const float* w = (const float*)d_in[1];   // weight [64,2048] f32

    const int H = in_sizes[1] / NEXPERT;      // 2048
    const int T = in_sizes[0] / H;            // 16384

    // d_out = topk_idx(int32 T*8) ++ topk_weight(f32 T*8) ++ row_idx(int32 T*8)
    int*   out_idx = (int*)d_out;
    float* out_w   = (float*)d_out + (size_t)T * TOPK;
    int*   out_row = (int*)d_out + 2 * (size_t)T * TOPK;

    dim3 grid(T / TOK_PER_BLK);
    dim3 block(THREADS);
    if (H == 2048) {
        hipLaunchKernelGGL(MoEGate_10187662426952_kernel<2048>, grid, block, 0,
                           stream, x, w, out_idx, out_w, out_row, T);
    } else {
        hipLaunchKernelGGL(MoEGate_10187662426952_kernel_dyn, grid, block, 0,
                           stream, x, w, out_idx, out_w, out_row, T, H);
    }
}